// RWW_Layer_83124797047473
// MI455X (gfx1250) — compile-verified
//
#include <hip/hip_runtime.h>

#define NREG   512
#define NSTEPS 20000
#define NB     32     // workgroups (all co-resident; grid << #WGPs)
#define RPB    16     // regions (matrix rows) per workgroup
#define TPB    256    // 8 waves of 32
#define NWAVES 8

typedef __attribute__((ext_vector_type(2))) float v2f;
typedef __attribute__((ext_vector_type(8))) float v8f;

__device__ __forceinline__ float Hfn(float I, float a, float b, float d, float bigc) {
    float x     = fmaf(a, I, -b);
    float numer = fabsf(x) + 1e-9f;
    float neg   = -d * x;
    float denom;
    if (neg > 50.0f) {
        denom = fabsf(1.0f - bigc * neg) + 1e-9f * d;
    } else {
        denom = fabsf(1.0f - expf(fminf(neg, 51.0f))) + 1e-9f * d;
    }
    return numer / denom;
}

// Re-initialize the cross-block barrier counter and S_E exchange buffer 0
// every call (d_ws is NOT re-poisoned between replays -> must be deterministic).
__global__ __launch_bounds__(TPB) void rww_init(const float* __restrict__ init_state,
                                                float* buf0, unsigned* cnt) {
    int i = blockIdx.x * blockDim.x + threadIdx.x;
    if (i < NREG) buf0[i] = init_state[2 * i];   // S_E column
    if (i == 0)   *cnt = 0u;
}

__global__ __launch_bounds__(TPB) void rww_main(const float* __restrict__ init_state,
                                                const float* __restrict__ Con,
                                                const float* __restrict__ vT,
                                                float* __restrict__ out,
                                                float* bufs,      // [2][NREG] S_E exchange
                                                unsigned* cnt) {  // barrier counter
    __shared__ __align__(16) float ldsA[RPB * NREG];    // 32 KB: Con slice, WMMA A-layout
    __shared__ __align__(16) float sE[NREG];            // shared S_E(t)
    __shared__ __align__(16) float part[NWAVES][RPB];   // per-wave GEMV partials

    const int tid  = threadIdx.x;
    const int blk  = blockIdx.x;
    const int lane = tid & 31;
    const int wave = tid >> 5;
    const int row0 = blk * RPB;

    // ---- Stage Con rows [row0, row0+16) into the 16x4-chunk WMMA A layout, ONCE ----
    // A frag (16x4 f32): VGPR0 holds K=base+0 (lanes 0-15) / K=base+2 (lanes 16-31),
    //                    VGPR1 holds K=base+1 / K=base+3.  One ds b64 load per chunk.
    for (int idx = tid; idx < RPB * NREG; idx += TPB) {
        int   m   = idx >> 9;                 // 0..15  (row within slice)
        int   k   = idx & (NREG - 1);         // 0..511 (column)
        float val = Con[(size_t)(row0 + m) * NREG + k];
        int chunk = k >> 2;
        int l     = m + (((k >> 1) & 1) << 4);
        int pos   = k & 1;
        ldsA[chunk * 64 + l * 2 + pos] = val;
    }

    // Per-region state lives in registers of threads 0..15 (one region each).
    float myE = 0.f, myI = 0.f, vCur = 0.f;
    if (tid < RPB) {
        myE  = init_state[2 * (row0 + tid) + 0];
        myI  = init_state[2 * (row0 + tid) + 1];
        vCur = vT[row0 + tid];                // noise for step 0, prefetched
    }
    __syncthreads();

    float* hist = out + 2 * NREG;   // state_hist follows state_vals in d_out

    const int cbase = wave * 16;
    const int koff  = (lane >= 16) ? 2 : 0;

    for (int step = 0; step < NSTEPS; ++step) {
        float* bufCur = bufs + ((step    ) & 1) * NREG;
        float* bufNxt = bufs + ((step + 1) & 1) * NREG;

        // ---- Pull S_E(t) from L2 (agent scope: skip stale WGP$) into LDS ----
        for (int i = tid; i < NREG; i += TPB) {
            sE[i] = __hip_atomic_load(&bufCur[i], __ATOMIC_RELAXED,
                                      __HIP_MEMORY_SCOPE_AGENT);
        }
        __syncthreads();

        // ---- GEMV slice via V_WMMA_F32_16X16X4_F32 (exact fp32 math) ----
        // Software-pipelined: issue all 32 ds_load_b64 into independent regs
        // (one LDS burst), then run 16 accumulating WMMAs back-to-back.
        v2f af[16], bf[16];
        #pragma unroll
        for (int cc = 0; cc < 16; ++cc) {
            int chunk = cbase + cc;
            af[cc] = *(const v2f*)&ldsA[chunk * 64 + lane * 2];
            bf[cc] = *(const v2f*)&sE[chunk * 4 + koff];
        }
        v8f c = {};
        #pragma unroll
        for (int cc = 0; cc < 16; ++cc) {
            c = __builtin_amdgcn_wmma_f32_16x16x4_f32(
                    false, af[cc], false, bf[cc], (short)0, c, false, false);
        }
        // D layout: VGPR j = row j (lanes 0-15) / row 8+j (lanes 16-31); cols identical.
        if (lane == 0) {
            #pragma unroll
            for (int j = 0; j < 8; ++j) part[wave][j] = c[j];
        } else if (lane == 16) {
            #pragma unroll
            for (int j = 0; j < 8; ++j) part[wave][8 + j] = c[j];
        }
        __syncthreads();

        // ---- Elementwise RWW update (one thread per owned region) ----
        if (tid < RPB) {
            float net = 0.f;
            #pragma unroll
            for (int w = 0; w < NWAVES; ++w) net += part[w][tid];

            int region = row0 + tid;

            // W_E*I_0=0.382, W_PLUS*J_NMDA=0.21, G*J_NMDA=3.0, W_I*I_0=0.2674, LAMBDA=0
            float I_E = 0.382f  + 0.21f * myE + 3.0f * net - myI;
            float I_I = 0.2674f + 0.15f * myE - myI;
            float rE  = Hfn(I_E, 310.0f, 125.0f, 0.16f,  1.0e9f);
            float rI  = Hfn(I_I, 615.0f, 177.0f, 0.087f, 1.0e5f);
            float dE  = -myE * 0.01f + (1.0f - myE) * 0.000641f * rE + 0.01f * vCur;
            float dI  = -myI * 0.1f  + 0.001f * rI                  + 0.01f * vCur;
            myE = fmaf(1e-4f, dE, myE);
            myI = fmaf(1e-4f, dI, myI);

            // Publish S_E(t+1) first: it is the only store the barrier must cover.
            __hip_atomic_store(&bufNxt[region], myE, __ATOMIC_RELAXED,
                               __HIP_MEMORY_SCOPE_AGENT);

            // One b64 non-temporal history store (keep L2 for v_of_T).
            size_t h = (size_t)step * (2 * NREG) + 2 * (size_t)region;
            union { float f[2]; double d; } hv;
            hv.f[0] = myE; hv.f[1] = myI;
            __builtin_nontemporal_store(hv.d, (double*)&hist[h]);

            // Prefetch next step's noise: its L2 latency hides under the barrier.
            if (step + 1 < NSTEPS) {
                vCur = vT[(size_t)(step + 1) * NREG + region];
            }
        }

        // ---- One device-wide barrier per step (monotonic counter, no reset race) ----
        __threadfence();      // make S_E publishes visible at device scope
        __syncthreads();
        if (tid == 0) {
            __hip_atomic_fetch_add(cnt, 1u, __ATOMIC_RELAXED,
                                   __HIP_MEMORY_SCOPE_AGENT);
            unsigned target = (unsigned)NB * (unsigned)(step + 1);
            while (__hip_atomic_load(cnt, __ATOMIC_RELAXED,
                                     __HIP_MEMORY_SCOPE_AGENT) < target) {
                __builtin_amdgcn_s_sleep(1);
            }
        }
        __syncthreads();
    }

    if (tid < RPB) {
        int region = row0 + tid;
        union { float f[2]; double d; } fv;
        fv.f[0] = myE; fv.f[1] = myI;
        *(double*)&out[2 * region] = fv.d;   // state_vals (final state)
    }
}

extern "C" void kernel_launch(void* const* d_in, const int* in_sizes, int n_in,
                              void* d_out, int out_size, void* d_ws, size_t ws_size,
                              hipStream_t stream) {
    const float* init_state = (const float*)d_in[0];   // (512, 2) f32
    const float* Con        = (const float*)d_in[1];   // (512, 512) f32
    const float* vT         = (const float*)d_in[2];   // (20000, 512) f32
    float*       out        = (float*)d_out;           // 1024 + 20480000 f32

    float*    bufs = (float*)d_ws;                       // 2 x 512 f32 S_E exchange
    unsigned* cnt  = (unsigned*)((char*)d_ws + 4096);    // barrier counter

    rww_init<<<2, TPB, 0, stream>>>(init_state, bufs, cnt);
    rww_main<<<NB, TPB, 0, stream>>>(init_state, Con, vT, out, bufs, cnt);
    (void)in_sizes; (void)n_in; (void)out_size; (void)ws_size;
}